// CoStGcnBlock_55001351192895
// MI455X (gfx1250) — compile-verified
//
#include <hip/hip_runtime.h>

// ---------------------------------------------------------------------------
// CoStGcnBlock for MI455X (gfx1250): fp32 path via V_WMMA_F32_16X16X4_F32.
//   Kernel 1: graph conv (adjacency WMMA + K=192 channel-mix WMMA) + BN1
//             + residual + ReLU -> y  (NCHW layout, in d_ws)
//   Kernel 2: temporal (9,1) conv as K=576 WMMA GEMM (im2col via LDS column
//             offset) + BN2 + bias + residual + ReLU -> out (NCHW)
// Tile staging uses GLOBAL_LOAD_ASYNC_TO_LDS_B128 (ASYNCcnt) when available.
// ---------------------------------------------------------------------------

#define N_ 32
#define C_ 64
#define O_ 64
#define T_ 1024
#define V_ 25
#define S_ 3
#define EPS_ 1e-5f

typedef float v2f __attribute__((ext_vector_type(2)));
typedef float v8f __attribute__((ext_vector_type(8)));
typedef int v4i __attribute__((vector_size(4 * sizeof(int))));

#if __has_builtin(__builtin_amdgcn_global_load_async_to_lds_b128)
#define HAS_ASYNC_LDS 1
#else
#define HAS_ASYNC_LDS 0
#endif

#if HAS_ASYNC_LDS
// 16B global -> LDS async copy (per-lane addresses), tracked on ASYNCcnt.
__device__ __forceinline__ void async_cp16(const float* g, float* l) {
  __builtin_amdgcn_global_load_async_to_lds_b128(
      (__attribute__((address_space(1))) v4i*)g,
      (__attribute__((address_space(3))) v4i*)l, /*offset=*/0, /*cpol=*/0);
}
#endif

__device__ __forceinline__ void wait_async0() {
#if HAS_ASYNC_LDS
#if __has_builtin(__builtin_amdgcn_s_wait_asynccnt)
  __builtin_amdgcn_s_wait_asynccnt(0);
#else
  asm volatile("s_wait_asynccnt 0" ::: "memory");
#endif
#endif
}

// D = A(16x4) * B(4x16) + C, all fp32, wave32.
// Assumed fragment layouts (ISA 7.12.2):
//   A: lanes 0-15 hold M=lane, VGPR0=K0,VGPR1=K1; lanes 16-31: K2,K3.
//   B: VGPR0 holds rows K0 (lanes 0-15) / K1 (lanes 16-31); VGPR1: K2/K3.
//   C/D: VGPR r -> M=r (lanes 0-15), M=r+8 (lanes 16-31); N = lane&15.
__device__ __forceinline__ v8f wmma_f32(v2f a, v2f b, v8f c) {
  return __builtin_amdgcn_wmma_f32_16x16x4_f32(
      /*neg_a=*/false, a, /*neg_b=*/false, b,
      /*c_mod=*/(short)0, c, /*reuse_a=*/false, /*reuse_b=*/false);
}

// ======================= Kernel 1: graph stage =============================
#define TB1 8
#define P1 (TB1 * V_)     // 200 positions per t-slab
#define XSW 204           // Xs row width: 200 valid + 4 zero (A-frag overrun)
#define P1T 13            // ceil(200/16) position tiles
#define P1PAD (P1T * 16)  // 208
#define KG (S_ * C_)      // 192 stacked (subset, channel) contraction

__global__ __launch_bounds__(256) void gcn_graph_kernel(
    const float* __restrict__ xg, const float* __restrict__ Ag,
    const float* __restrict__ attng, const float* __restrict__ wgg,
    const float* __restrict__ bgg, const float* __restrict__ g1,
    const float* __restrict__ b1, const float* __restrict__ m1,
    const float* __restrict__ v1, float* __restrict__ yg) {
  __shared__ float Xs[C_][XSW];       // x tile [c][t*25+v]          (52.2 KB)
  __shared__ float XP[KG][P1PAD];     // adjacency output, stacked s (159.7 KB)
  __shared__ float Wg[O_][KG];        // channel-mix weights         (49.2 KB)
  __shared__ float Awp[S_][28][32];   // zero-padded A*attn          (10.8 KB)
  __shared__ float sc1[O_], sh1[O_];

  const int tid = threadIdx.x;
  const int wave = tid >> 5;
  const int lane = tid & 31;
  const int l16 = lane & 15;
  const int hi = lane >> 4;
  const int n = blockIdx.y;
  const int t0 = blockIdx.x * TB1;

  // ---------------- stage loads ----------------
#if HAS_ASYNC_LDS
  // x tile: per channel c, 200 contiguous floats (t0..t0+7, all v) = 50 x 16B
  for (int i = tid; i < C_ * 50; i += 256) {
    int c = i / 50, j = i % 50;
    async_cp16(xg + ((n * C_ + c) * T_ + t0) * V_ + j * 4, &Xs[c][j * 4]);
  }
  // Wg[o][s*64+c]: per (s,o), 64 contiguous floats = 16 x 16B
  for (int i = tid; i < S_ * O_ * 16; i += 256) {
    int j = i & 15, so = i >> 4;
    int s = so >> 6, o = so & 63;
    async_cp16(wgg + (s * O_ + o) * C_ + j * 4, &Wg[o][s * C_ + j * 4]);
  }
#else
  for (int i = tid; i < C_ * P1; i += 256) {
    int c = i / P1, pos = i % P1;
    int t = pos / V_, v = pos % V_;
    Xs[c][pos] = xg[((n * C_ + c) * T_ + t0 + t) * V_ + v];
  }
  for (int i = tid; i < O_ * KG; i += 256) {
    int o = i / KG, k = i % KG;
    int s = k >> 6, c = k & 63;
    Wg[o][k] = wgg[(s * O_ + o) * C_ + c];
  }
#endif
  // zero Xs tail columns [200,204) so unguarded A-frag reads stay in-bounds
  for (int i = tid; i < C_ * (XSW - P1); i += 256) {
    Xs[i / (XSW - P1)][P1 + i % (XSW - P1)] = 0.f;
  }
  for (int i = tid; i < S_ * 28 * 32; i += 256) {
    int s = i / (28 * 32), r = i % (28 * 32);
    int w = r / 32, v = r % 32;
    float val = 0.f;
    if (w < V_ && v < V_) {
      int idx = (s * V_ + w) * V_ + v;
      val = Ag[idx] * attng[idx];
    }
    Awp[s][w][v] = val;
  }
  for (int i = tid; i < KG * (P1PAD - P1); i += 256) {  // zero pad columns
    int r = i / (P1PAD - P1), cc = P1 + i % (P1PAD - P1);
    XP[r][cc] = 0.f;
  }
  if (tid < O_) {
    float inv = g1[tid] * rsqrtf(v1[tid] + EPS_);
    float bsum = bgg[tid] + bgg[O_ + tid] + bgg[2 * O_ + tid];
    sc1[tid] = inv;
    sh1[tid] = bsum * inv + (b1[tid] - m1[tid] * inv);
  }
  wait_async0();
  __syncthreads();

  // ---- phase 1: adjacency, per subset: XP[s*64+c][(t,v)] = X[(c,t),w]*Awp
  // M = 512 rows (c-major (c,t)), K = 25 pad 28, N = 25 pad 32.
  // Unguarded A reads: w in [25,28) hit zero-padded / don't-care lanes whose
  // B rows (Awp[w>=25]) are zero, so products vanish.
  for (int tIdx = wave; tIdx < S_ * 32 * 2; tIdx += 8) {
    int nt = tIdx & 1;
    int mt = (tIdx >> 1) & 31;
    int s = tIdx >> 6;
    int m0 = mt * 16, v0 = nt * 16;
    int mrow = m0 + l16;
    int c = mrow >> 3, t = mrow & 7;
    v8f acc = {0.f, 0.f, 0.f, 0.f, 0.f, 0.f, 0.f, 0.f};
#pragma unroll
    for (int kk = 0; kk < 28; kk += 4) {
      v2f a, b;
      int w0 = kk + 2 * hi;
      a.x = Xs[c][t * V_ + w0];
      a.y = Xs[c][t * V_ + w0 + 1];
      b.x = Awp[s][kk + hi][v0 + l16];
      b.y = Awp[s][kk + hi + 2][v0 + l16];
      acc = wmma_f32(a, b, acc);
    }
#pragma unroll
    for (int r = 0; r < 8; ++r) {
      int mm = m0 + r + 8 * hi;
      int v = v0 + l16;
      if (v < V_) {
        int cc = mm >> 3, tt = mm & 7;
        XP[s * C_ + cc][tt * V_ + v] = acc[r];
      }
    }
  }
  __syncthreads();

  // ---- phase 2: D[o,pos] = Wg[o][0..191] . XP[0..191][pos], fused epilogue
  for (int tIdx = wave; tIdx < 4 * P1T; tIdx += 8) {
    int o0 = (tIdx / P1T) * 16;
    int p0 = (tIdx % P1T) * 16;
    v8f acc = {0.f, 0.f, 0.f, 0.f, 0.f, 0.f, 0.f, 0.f};
#pragma unroll 4
    for (int kk = 0; kk < KG; kk += 4) {
      v2f a, b;
      a.x = Wg[o0 + l16][kk + 2 * hi];
      a.y = Wg[o0 + l16][kk + 2 * hi + 1];
      b.x = XP[kk + hi][p0 + l16];
      b.y = XP[kk + hi + 2][p0 + l16];
      acc = wmma_f32(a, b, acc);
    }
#pragma unroll
    for (int r = 0; r < 8; ++r) {
      int o = o0 + r + 8 * hi;
      int pos = p0 + l16;
      if (pos < P1) {
        int t = pos / V_, v = pos % V_;
        float val = acc[r] * sc1[o] + sh1[o] + Xs[o][pos];  // BN1 + residual
        yg[((n * O_ + o) * T_ + t0 + t) * V_ + v] = fmaxf(val, 0.f);
      }
    }
  }
}

// ======================= Kernel 2: temporal stage ==========================
#define TB2 8
#define P2 (TB2 * V_)  // 200 output positions
#define P2T 13
#define HALO_ 4
#define TIN 16   // TB2 + 2*HALO
#define YW 416   // 16*25 valid cols + zero pad (max read col = 407)
#define KT 576   // 9 taps * 64 channels, k-major: kk = k*64 + c

__global__ __launch_bounds__(256) void tcn_kernel(
    const float* __restrict__ yg, const float* __restrict__ wtg,
    const float* __restrict__ btg, const float* __restrict__ g2,
    const float* __restrict__ b2, const float* __restrict__ m2,
    const float* __restrict__ v2, const float* __restrict__ xg,
    float* __restrict__ outg) {
  __shared__ float Yt[C_][YW];   // y tile with temporal halo (106.5 KB)
  __shared__ float Wt[O_][KT];   // temporal weights           (147.5 KB)
  __shared__ float sc2[O_], sh2[O_];

  const int tid = threadIdx.x;
  const int wave = tid >> 5;
  const int lane = tid & 31;
  const int l16 = lane & 15;
  const int hi = lane >> 4;
  const int n = blockIdx.y;
  const int t0 = blockIdx.x * TB2;

  // y tile: input t' = t0-4 .. t0+11, zero-filled outside [0,T)
  const int gtS = (t0 - HALO_ < 0) ? 0 : t0 - HALO_;
  const int gtE = (t0 + TB2 + HALO_ > T_) ? T_ : t0 + TB2 + HALO_;
  const int nrows = gtE - gtS;               // 12 (edges) or 16
  const int colS = (gtS - (t0 - HALO_)) * V_;  // 0 or 100 (16B aligned)
#if HAS_ASYNC_LDS
  {
    const int cpr = nrows * V_ / 4;  // 16B chunks per channel row (75 or 100)
    for (int i = tid; i < C_ * cpr; i += 256) {
      int c = i / cpr, j = i % cpr;
      async_cp16(yg + ((n * C_ + c) * T_ + gtS) * V_ + j * 4,
                 &Yt[c][colS + j * 4]);
    }
  }
#else
  for (int i = tid; i < C_ * nrows * V_; i += 256) {
    int c = i / (nrows * V_), r = i % (nrows * V_);
    Yt[c][colS + r] = yg[((n * C_ + c) * T_ + gtS) * V_ + r];
  }
#endif
  // zero-fill halo-out-of-range cols + pad tail [colS+nrows*25 .. YW)
  {
    const int padLo = colS;
    const int hiStart = colS + nrows * V_;
    const int padHi = YW - hiStart;
    const int padTot = padLo + padHi;
    for (int i = tid; i < C_ * padTot; i += 256) {
      int c = i / padTot, p = i % padTot;
      int col = (p < padLo) ? p : (hiStart + p - padLo);
      Yt[c][col] = 0.f;
    }
  }
  // Wt reorder (stride-9 gather, stays on VGPR path): kk = k*64 + c
  for (int i = tid; i < O_ * KT; i += 256) {
    int o = i / KT, kk = i % KT;
    int k = kk >> 6, c = kk & 63;
    Wt[o][kk] = wtg[(o * C_ + c) * 9 + k];  // wt (O, C, 9, 1)
  }
  if (tid < O_) {
    float inv = g2[tid] * rsqrtf(v2[tid] + EPS_);
    sc2[tid] = inv;
    sh2[tid] = btg[tid] * inv + (b2[tid] - m2[tid] * inv);
  }
  wait_async0();
  __syncthreads();

  // D[o,pos] = sum_kk Wt[o][kk] * y[c=kk&63][t + (kk>>6)] ; im2col = +k*25 col
  for (int tIdx = wave; tIdx < 4 * P2T; tIdx += 8) {
    int o0 = (tIdx / P2T) * 16;
    int p0 = (tIdx % P2T) * 16;
    int pcol = p0 + l16;
    v8f acc = {0.f, 0.f, 0.f, 0.f, 0.f, 0.f, 0.f, 0.f};
#pragma unroll 4
    for (int kk = 0; kk < KT; kk += 4) {
      v2f a, b;
      a.x = Wt[o0 + l16][kk + 2 * hi];
      a.y = Wt[o0 + l16][kk + 2 * hi + 1];
      int kr0 = kk + hi;      // B row for VGPR0
      int kr1 = kk + hi + 2;  // B row for VGPR1
      b.x = Yt[kr0 & 63][pcol + (kr0 >> 6) * V_];
      b.y = Yt[kr1 & 63][pcol + (kr1 >> 6) * V_];
      acc = wmma_f32(a, b, acc);
    }
#pragma unroll
    for (int r = 0; r < 8; ++r) {
      int o = o0 + r + 8 * hi;
      int pos = p0 + l16;
      if (pos < P2) {
        int t = pos / V_, v = pos % V_;
        int idx = ((n * O_ + o) * T_ + t0 + t) * V_ + v;
        float val = acc[r] * sc2[o] + sh2[o] + xg[idx];  // BN2+bias+residual
        outg[idx] = fmaxf(val, 0.f);
      }
    }
  }
}

// ============================ launcher =====================================
extern "C" void kernel_launch(void* const* d_in, const int* in_sizes, int n_in,
                              void* d_out, int out_size, void* d_ws,
                              size_t ws_size, hipStream_t stream) {
  (void)in_sizes; (void)n_in; (void)out_size; (void)ws_size;
  const float* x    = (const float*)d_in[0];
  const float* A    = (const float*)d_in[1];
  const float* attn = (const float*)d_in[2];
  const float* wg   = (const float*)d_in[3];
  const float* bg   = (const float*)d_in[4];
  const float* g1   = (const float*)d_in[5];
  const float* b1   = (const float*)d_in[6];
  const float* m1   = (const float*)d_in[7];
  const float* v1   = (const float*)d_in[8];
  const float* wt   = (const float*)d_in[9];
  const float* bt   = (const float*)d_in[10];
  const float* g2   = (const float*)d_in[11];
  const float* b2   = (const float*)d_in[12];
  const float* m2   = (const float*)d_in[13];
  const float* v2   = (const float*)d_in[14];
  float* out = (float*)d_out;
  float* y = (float*)d_ws;  // N*O*T*V floats = 210 MB intermediate

  dim3 grid1(T_ / TB1, N_);
  gcn_graph_kernel<<<grid1, 256, 0, stream>>>(x, A, attn, wg, bg, g1, b1, m1,
                                              v1, y);
  dim3 grid2(T_ / TB2, N_);
  tcn_kernel<<<grid2, 256, 0, stream>>>(y, wt, bt, g2, b2, m2, v2, x, out);
}